// ScaleDegreeChordModelForExport_29695403885230
// MI455X (gfx1250) — compile-verified
//
#include <hip/hip_runtime.h>
#include <hip/hip_bf16.h>
#include <math.h>

typedef float v2f __attribute__((ext_vector_type(2)));
typedef float v8f __attribute__((ext_vector_type(8)));

#define NBINS  216
#define CDIM   13
#define KSTEPS 54            // 216 / 4
#define WMMA4(A,B,C) __builtin_amdgcn_wmma_f32_16x16x4_f32(false,(A),false,(B),(short)0,(C),false,false)

__device__ __forceinline__ float lane_read(float v, int srcLane) {
    // ds_bpermute_b32: dst[lane] = src[srcLane & 31] (byte index)
    return __builtin_bit_cast(float,
        __builtin_amdgcn_ds_bpermute(srcLane << 2, __builtin_bit_cast(int, v)));
}

// softmax over N (cols 0..12) for an f32 16x16 C/D fragment.
// lane holds col c = lane&15 of rows r (+8 if lane>=16), r = 0..7.
__device__ __forceinline__ void softmax16(v8f& acc, float bias, int c, int lane) {
    #pragma unroll
    for (int r = 0; r < 8; ++r) {
        float lg = acc[r] + bias;
        float m = (c < CDIM) ? lg : -1e30f;
        #pragma unroll
        for (int off = 8; off; off >>= 1)
            m = fmaxf(m, lane_read(m, lane ^ off));           // stays inside 16-lane half
        float e = (c < CDIM) ? __expf(lg - m) : 0.0f;
        float s = e;
        #pragma unroll
        for (int off = 8; off; off >>= 1)
            s += lane_read(s, lane ^ off);
        acc[r] = e / s;
    }
}

__global__ __launch_bounds__(256)
void chord_heads_kernel(const float* __restrict__ x,
                        const float* __restrict__ Wk, const float* __restrict__ bk,
                        const float* __restrict__ Wd, const float* __restrict__ bd,
                        const float* __restrict__ Wb, const float* __restrict__ bb,
                        float* __restrict__ out, int R)
{
    // --- stage WMMA B-fragments (3 heads x 54 K-steps) into LDS, zero-padded to N=16 ---
    __shared__ v2f wfrag[3][KSTEPS][32];          // 41472 bytes
    const int tid = threadIdx.x;
    for (int idx = tid; idx < 3 * KSTEPS * 32; idx += blockDim.x) {
        int h   = idx / (KSTEPS * 32);
        int rem = idx - h * (KSTEPS * 32);
        int s   = rem >> 5;
        int l   = rem & 31;
        int cc  = l & 15;
        int hh  = l >> 4;
        int k0  = 4 * s + 2 * hh;                 // max 214; k0+1 <= 215 < 216
        const float* W = (h == 0) ? Wk : ((h == 1) ? Wd : Wb);
        v2f w; w.x = 0.0f; w.y = 0.0f;
        if (cc < CDIM) { w.x = W[k0 * CDIM + cc]; w.y = W[(k0 + 1) * CDIM + cc]; }
        wfrag[h][s][l] = w;
    }
    __syncthreads();

    const int lane = tid & 31;
    const int wave = tid >> 5;
    const int c    = lane & 15;                   // N column owned by this lane
    const int hi   = lane >> 4;
    const int base = lane & 16;                   // first lane of my 16-lane half

    const float biasK = (c < CDIM) ? bk[c] : 0.0f;
    const float biasD = (c < CDIM) ? bd[c] : 0.0f;
    const float biasB = (c < CDIM) ? bb[c] : 0.0f;

    const int tiles = (R + 15) >> 4;
    const int gw    = blockIdx.x * (blockDim.x >> 5) + wave;
    const int nw    = gridDim.x * (blockDim.x >> 5);
    const long long hstride = (long long)R * CDIM;

    for (int t = gw; t < tiles; t += nw) {
        const int row0 = t << 4;
        int rl = row0 + (lane & 15);
        int rc = rl < R ? rl : R - 1;
        const float* xrow = x + (long long)rc * NBINS + 2 * hi;

        // prefetch next tile's rows while we crunch this one
        int tn = t + nw;
        if (tn < tiles) {
            int rn = (tn << 4) + (lane & 15);
            rn = rn < R ? rn : R - 1;
            __builtin_prefetch(x + (long long)rn * NBINS, 0, 1);
        }

        v8f accK = {0.f,0.f,0.f,0.f,0.f,0.f,0.f,0.f};
        v8f accD = {0.f,0.f,0.f,0.f,0.f,0.f,0.f,0.f};
        v8f accB = {0.f,0.f,0.f,0.f,0.f,0.f,0.f,0.f};

        // software-pipelined K loop: 1 global b64 + 3 ds b64 + 3 WMMA per step
        v2f a  = *(const v2f*)(xrow);
        v2f w0 = wfrag[0][0][lane];
        v2f w1 = wfrag[1][0][lane];
        v2f w2 = wfrag[2][0][lane];
        #pragma unroll 3
        for (int s = 0; s < KSTEPS - 1; ++s) {
            v2f a_n  = *(const v2f*)(xrow + 4 * (s + 1));
            v2f w0n = wfrag[0][s + 1][lane];
            v2f w1n = wfrag[1][s + 1][lane];
            v2f w2n = wfrag[2][s + 1][lane];
            accK = WMMA4(a, w0, accK);
            accD = WMMA4(a, w1, accD);
            accB = WMMA4(a, w2, accB);
            a = a_n; w0 = w0n; w1 = w1n; w2 = w2n;
        }
        accK = WMMA4(a, w0, accK);
        accD = WMMA4(a, w1, accD);
        accB = WMMA4(a, w2, accB);

        // --- epilogue: bias + softmax (in place), all lanes active ---
        softmax16(accK, biasK, c, lane);
        softmax16(accD, biasD, c, lane);
        softmax16(accB, biasB, c, lane);

        // --- store key/degree/bass probs ---
        #pragma unroll
        for (int r = 0; r < 8; ++r) {
            int row = row0 + r + 8 * hi;
            if (row < R && c < CDIM) {
                long long o = (long long)row * CDIM + c;
                out[o]                = accK[r];
                out[hstride + o]      = accD[r];
                out[2 * hstride + o]  = accB[r];
            }
        }

        // --- root head: root0 = k0+d0-k0*d0 ; root_pc[r] = sum_k key[k+1]*deg[((r-k)%12)+1]
        #pragma unroll
        for (int r = 0; r < 8; ++r) {
            float k0v = lane_read(accK[r], base);      // col 0 of my half
            float d0v = lane_read(accD[r], base);
            float root0 = k0v + d0v - k0v * d0v;
            int rr = c - 1;                            // my root pitch-class (valid for c in 1..12)
            float accR = 0.0f;
            #pragma unroll
            for (int k = 0; k < 12; ++k) {
                float kk = lane_read(accK[r], base + k + 1);
                int di = rr - k; di += (di < 0) ? 12 : 0; di += (di < 0) ? 12 : 0;
                float dd = lane_read(accD[r], base + di + 1);
                accR = fmaf(kk, dd, accR);
            }
            float val = (c == 0) ? root0 : accR;
            int row = row0 + r + 8 * hi;
            if (row < R && c < CDIM)
                out[3 * hstride + (long long)row * CDIM + c] = val;
        }
    }
}

extern "C" void kernel_launch(void* const* d_in, const int* in_sizes, int n_in,
                              void* d_out, int out_size, void* d_ws, size_t ws_size,
                              hipStream_t stream) {
    const float* x  = (const float*)d_in[0];
    const float* Wk = (const float*)d_in[1];
    const float* bk = (const float*)d_in[2];
    const float* Wd = (const float*)d_in[3];
    const float* bd = (const float*)d_in[4];
    const float* Wb = (const float*)d_in[5];
    const float* bb = (const float*)d_in[6];
    float* out = (float*)d_out;
    int R = in_sizes[0] / NBINS;                 // 262144 for the reference shapes

    dim3 block(256);                             // 8 wave32 waves
    dim3 grid(512);                              // 4096 waves -> 4 tiles each
    chord_heads_kernel<<<grid, block, 0, stream>>>(x, Wk, bk, Wd, bd, Wb, bb, out, R);
}